// Model_Base_6691559047734
// MI455X (gfx1250) — compile-verified
//
#include <hip/hip_runtime.h>

typedef __attribute__((ext_vector_type(2))) float v2f;
typedef __attribute__((ext_vector_type(8))) float v8f;

#define BATCH 32
#define NROWS 1024
#define DDIM  256
#define SPLIT 8

// ---------------------------------------------------------------------------
// Kernel 1: partial column-sums.  sx[b,d] = sum_n x[b,n,d]; same for y.
// Thread t owns column d=t -> consecutive lanes read consecutive floats
// (perfectly coalesced 128B per wave-iteration).
// ---------------------------------------------------------------------------
__global__ __launch_bounds__(256) void reduce_partial(const float* __restrict__ x,
                                                      const float* __restrict__ y,
                                                      float* __restrict__ partial) {
    const int t     = threadIdx.x;
    const int chunk = blockIdx.x;   // which block of 128 rows
    const int which = blockIdx.y;   // 0 = x, 1 = y
    const int b     = blockIdx.z;
    const float* src = which ? y : x;
    const float* p = src + ((size_t)b * NROWS + (size_t)chunk * (NROWS / SPLIT)) * DDIM + t;
    float acc = 0.0f;
#pragma unroll 8
    for (int r = 0; r < NROWS / SPLIT; ++r)
        acc += p[(size_t)r * DDIM];
    partial[(((size_t)which * BATCH + b) * SPLIT + chunk) * DDIM + t] = acc;
}

// ---------------------------------------------------------------------------
// Kernel 2: fold the SPLIT partials -> sums[2][BATCH][DDIM]
//   sums[0] = sx (from x), sums[1] = sy (from y)
// ---------------------------------------------------------------------------
__global__ __launch_bounds__(256) void reduce_combine(const float* __restrict__ partial,
                                                      float* __restrict__ sums) {
    const int t     = threadIdx.x;
    const int b     = blockIdx.x;
    const int which = blockIdx.y;
    float acc = 0.0f;
#pragma unroll
    for (int c = 0; c < SPLIT; ++c)
        acc += partial[(((size_t)which * BATCH + b) * SPLIT + c) * DDIM + t];
    sums[((size_t)which * BATCH + b) * DDIM + t] = acc;
}

// ---------------------------------------------------------------------------
// Kernel 3: batched 16-row dot products via V_WMMA_F32_16X16X4_F32.
//   h==0: rowsum[b,m] = sx[b] . y[b,m]   -> out[b*2048 + m]
//   h==1: colsum[b,n] = sy[b] . x[b,n]   -> out[b*2048 + 1024 + n]
// A (16x4 f32): lane L holds row M=L&15, K = 2*(L>>4) + {vgpr0,vgpr1}
// B (4x16 f32): value independent of N column (broadcast of sum vector),
//               loaded with the mirrored K mapping.
// D (16x16 f32): replicated across N; lane 0 carries M=0..7, lane 16 M=8..15.
// ---------------------------------------------------------------------------
__global__ __launch_bounds__(256) void dot_wmma(const float* __restrict__ x,
                                                const float* __restrict__ y,
                                                const float* __restrict__ sums,
                                                float* __restrict__ out) {
    const int wave = threadIdx.x >> 5;
    const int lane = threadIdx.x & 31;
    const int g    = blockIdx.x * 8 + wave;   // 0..4095 tiles of 16 outputs
    const int h    = g >> 11;                 // 0 = rowsum, 1 = colsum
    const int rem  = g & 2047;
    const int b    = rem >> 6;
    const int mbase = (rem & 63) << 4;

    const float* data = h ? x : y;                             // data matrix
    const float* vec  = sums + ((size_t)h * BATCH + b) * DDIM; // matching sum vector

    const int row  = mbase + (lane & 15);
    const int koff = (lane >> 4) << 1;                         // 0 or 2
    const float* arow = data + ((size_t)b * NROWS + row) * DDIM + koff;
    const float* brow = vec + koff;

    v8f c0 = {};
    v8f c1 = {};
#pragma unroll 4
    for (int d0 = 0; d0 < DDIM; d0 += 8) {
        v2f a0 = *(const v2f*)(arow + d0);
        v2f b0 = *(const v2f*)(brow + d0);
        c0 = __builtin_amdgcn_wmma_f32_16x16x4_f32(false, a0, false, b0,
                                                   (short)0, c0, false, false);
        v2f a1 = *(const v2f*)(arow + d0 + 4);
        v2f b1 = *(const v2f*)(brow + d0 + 4);
        c1 = __builtin_amdgcn_wmma_f32_16x16x4_f32(false, a1, false, b1,
                                                   (short)0, c1, false, false);
    }
    v8f c = c0 + c1;

    const size_t obase = (size_t)b * 2048 + (size_t)h * 1024 + (size_t)mbase;
    if (lane == 0) {
#pragma unroll
        for (int i = 0; i < 8; ++i) out[obase + i] = c[i];
    } else if (lane == 16) {
#pragma unroll
        for (int i = 0; i < 8; ++i) out[obase + 8 + i] = c[i];
    }
}

// ---------------------------------------------------------------------------
extern "C" void kernel_launch(void* const* d_in, const int* in_sizes, int n_in,
                              void* d_out, int out_size, void* d_ws, size_t ws_size,
                              hipStream_t stream) {
    const float* x = (const float*)d_in[0];   // [32,1024,256]
    const float* y = (const float*)d_in[1];   // [32,1024,256]
    float* out = (float*)d_out;               // [32,2048]

    float* partial = (float*)d_ws;                              // 2*32*8*256 f = 512 KB
    float* sums    = partial + (size_t)2 * BATCH * SPLIT * DDIM; // 2*32*256 f = 64 KB

    reduce_partial<<<dim3(SPLIT, 2, BATCH), 256, 0, stream>>>(x, y, partial);
    reduce_combine<<<dim3(BATCH, 2), 256, 0, stream>>>(partial, sums);
    dot_wmma<<<dim3(512), 256, 0, stream>>>(x, y, sums, out);
}